// LearnableSplineActivation_33294586478908
// MI455X (gfx1250) — compile-verified
//
#include <hip/hip_runtime.h>

// Uniform cubic B-spline activation, CDNA5 / gfx1250 (wave32).
// y = sum_j b_j(w) * c[i-3+j],  u = 69*x, i = clamp(floor(u),3,65), w = u - i.
// Memory-bound: 8 B/element -> ~23 us floor at 23.3 TB/s. Strategy:
//  - 128-bit non-temporal global loads/stores (streaming, no L2 pollution)
//  - coefficient staging via global_load_async_to_lds_b32 (ASYNCcnt path)
//  - LDS window table: one ds_load_b128 gather per element instead of 4x b32
//  - unguarded fast path for full tiles (kills per-iteration exec-mask SALU)
//  - global_prefetch_b8 toward the next tile

typedef float f4 __attribute__((ext_vector_type(4)));

#define BLOCK  256
#define UNROLL 8

__device__ __forceinline__ float spline_eval(float x, const f4* __restrict__ win) {
    float u  = x * 69.0f;
    float fi = floorf(u);
    fi = fminf(fmaxf(fi, 3.0f), 65.0f);       // -> v_med3_f32
    float w  = u - fi;
    int   k  = (int)fi - 3;                   // 0..62

    f4 d = win[k];                            // ds_load_b128 (16B aligned)

    float w2 = w * w;
    float w3 = w2 * w;
    float s  = 1.0f - w;
    float s3 = (s * s) * s;
    const float i6 = 0x1.555556p-3f;          // 1/6
    float b0 = s3 * i6;
    float b3 = w3 * i6;
    float b1 = fmaf(0.5f, w3, 0x1.555556p-1f) - w2;   // 0.5w^3 - w^2 + 2/3
    float b2 = 1.0f - b0 - b1 - b3;                   // partition of unity

    return fmaf(b3, d.w, fmaf(b2, d.z, fmaf(b1, d.y, b0 * d.x)));
}

__global__ __launch_bounds__(BLOCK)
void LearnableSplineActivation_33294586478908_kernel(const float* __restrict__ x,
                                                     const float* __restrict__ coef,
                                                     float* __restrict__ out,
                                                     long long n, int ncoef) {
    __shared__ float craw[68];   // raw coefficients, async-DMA'd from global
    __shared__ f4    win[64];    // win[k] = {c[k], c[k+1], c[k+2], c[k+3]}, k=0..62

    const int tid = threadIdx.x;

    // Zero-fill only slots NOT covered by the async DMA (no LDS write race).
    if (tid < 68 && tid >= ncoef) craw[tid] = 0.0f;

    // CDNA5 async memory->LDS copy, one coefficient per lane (ASYNCcnt path).
    if (tid < 67 && tid < ncoef) {
        unsigned     ldsOff = (unsigned)(uintptr_t)(&craw[tid]); // low 32b = LDS offset
        const float* gptr   = coef + tid;
        asm volatile("global_load_async_to_lds_b32 %0, %1, off"
                     :: "v"(ldsOff), "v"(gptr)
                     : "memory");
    }
    asm volatile("s_wait_asynccnt 0x0" ::: "memory");
    __syncthreads();

    // Build the windowed 16B-aligned gather table from LDS.
    if (tid < 63) {
        f4 v;
        v.x = craw[tid + 0];
        v.y = craw[tid + 1];
        v.z = craw[tid + 2];
        v.w = craw[tid + 3];
        win[tid] = v;
    }
    __syncthreads();

    const f4* __restrict__ xv = (const f4*)x;
    f4* __restrict__       ov = (f4*)out;
    const long long nvec    = n >> 2;
    const long long tileLen = (long long)BLOCK * UNROLL;
    const long long base    = (long long)blockIdx.x * tileLen + tid;
    const long long tileEnd = ((long long)blockIdx.x + 1) * tileLen;

    // Prefetch the start of the next tile into L2 (global_prefetch_b8).
    if (base + tileLen < nvec && (tid & 7) == 0)
        __builtin_prefetch((const void*)(xv + base + tileLen), 0, 0);

    if (tileEnd <= nvec) {
        // Full tile: no per-iteration bounds checks, no exec-mask churn.
#pragma unroll
        for (int it = 0; it < UNROLL; ++it) {
            const long long idx = base + (long long)it * BLOCK;
            f4 v = __builtin_nontemporal_load(&xv[idx]);   // global_load_b128 TH=NT
            f4 r;
            r.x = spline_eval(v.x, win);
            r.y = spline_eval(v.y, win);
            r.z = spline_eval(v.z, win);
            r.w = spline_eval(v.w, win);
            __builtin_nontemporal_store(r, &ov[idx]);      // global_store_b128 TH=NT
        }
    } else {
        // Partial (last) tile: guarded.
#pragma unroll
        for (int it = 0; it < UNROLL; ++it) {
            const long long idx = base + (long long)it * BLOCK;
            if (idx < nvec) {
                f4 v = __builtin_nontemporal_load(&xv[idx]);
                f4 r;
                r.x = spline_eval(v.x, win);
                r.y = spline_eval(v.y, win);
                r.z = spline_eval(v.z, win);
                r.w = spline_eval(v.w, win);
                __builtin_nontemporal_store(r, &ov[idx]);
            }
        }
        // Scalar tail (n % 4 == 0 for this problem, but stay correct in general).
        if (tid == 0 && tileEnd >= nvec) {
            for (long long kk = (nvec << 2); kk < n; ++kk)
                out[kk] = spline_eval(x[kk], win);
        }
    }
}

extern "C" void kernel_launch(void* const* d_in, const int* in_sizes, int n_in,
                              void* d_out, int out_size, void* d_ws, size_t ws_size,
                              hipStream_t stream) {
    const float* x    = (const float*)d_in[0];
    const float* coef = (const float*)d_in[1];
    float*       out  = (float*)d_out;
    const long long n     = (long long)in_sizes[0];
    const int       ncoef = in_sizes[1];

    const long long nvec     = n >> 2;
    const long long perBlock = (long long)BLOCK * UNROLL;
    long long blocks = (nvec + perBlock - 1) / perBlock;
    if (blocks < 1) blocks = 1;

    LearnableSplineActivation_33294586478908_kernel<<<(unsigned)blocks, BLOCK, 0, stream>>>(
        x, coef, out, n, ncoef);
}